// EcoPNAGNN_31542239822517
// MI455X (gfx1250) — compile-verified
//
#include <hip/hip_runtime.h>
#include <math.h>

typedef __attribute__((ext_vector_type(2))) float v2f;
typedef __attribute__((ext_vector_type(8))) float v8f;

#define F 64
#define NF 32
#define EF 16
#define KPRE 192
#define KPOST 832
#define EDGE_WAVES 4
#define POST_WAVES 4

__device__ __forceinline__ v8f wmma_f32(v2f a, v2f b, v8f c) {
  // D = A(16x4 f32) * B(4x16 f32) + C(16x16 f32)
  return __builtin_amdgcn_wmma_f32_16x16x4_f32(false, a, false, b, (short)0, c,
                                               false, false);
}

// order-preserving float -> uint key for atomicMin/Max
__device__ __forceinline__ unsigned fkey(float f) {
  unsigned u = __float_as_uint(f);
  return (u & 0x80000000u) ? ~u : (u | 0x80000000u);
}
__device__ __forceinline__ float funkey(unsigned u) {
  unsigned r = (u & 0x80000000u) ? (u ^ 0x80000000u) : ~u;
  return __uint_as_float(r);
}

// ---------------- small setup kernels ----------------
__global__ void k_zero_init(float* deg, float* avgsum, int N) {
  int i = blockIdx.x * blockDim.x + threadIdx.x;
  if (i < N) deg[i] = 0.f;
  if (i == 0) avgsum[0] = 0.f;
}

__global__ void k_degree(const int* dst, float* deg, int E) {
  int i = blockIdx.x * blockDim.x + threadIdx.x;
  if (i < E) atomicAdd(&deg[dst[i]], 1.f);
}

__global__ void k_avglog(const float* deg, float* avgsum, int N) {
  __shared__ float red[256];
  int i = blockIdx.x * blockDim.x + threadIdx.x;
  float v = (i < N) ? logf(deg[i] + 1.f) : 0.f;
  red[threadIdx.x] = v;
  __syncthreads();
  for (int o = 128; o > 0; o >>= 1) {
    if ((int)threadIdx.x < o) red[threadIdx.x] += red[threadIdx.x + o];
    __syncthreads();
  }
  if (threadIdx.x == 0) atomicAdd(avgsum, red[0]);
}

__global__ void k_reset_agg(float* sumb, unsigned* minb, unsigned* maxb,
                            float* sqb, int total) {
  int i = blockIdx.x * blockDim.x + threadIdx.x;
  if (i < total) {
    sumb[i] = 0.f;
    sqb[i]  = 0.f;
    minb[i] = 0xFFFFFFFFu;
    maxb[i] = 0u;
  }
}

// ---------------- encoder: h = relu(x @ nW.T + nb) ----------------
__global__ void k_encoder(const float* __restrict__ x,
                          const float* __restrict__ nW,
                          const float* __restrict__ nb,
                          float* __restrict__ h, int N) {
  int lane = threadIdx.x & 31, wv = threadIdx.x >> 5;
  int tile = blockIdx.x * (blockDim.x >> 5) + wv;
  int n0 = tile * 16;
  if (n0 >= N) return;                       // wave-uniform
  int row = lane & 15, half = lane >> 4;
  int n = min(n0 + row, N - 1);
  v8f acc[4] = {};
  for (int s = 0; s < NF / 4; ++s) {
    int k = 4 * s + 2 * half;
    float2 av = *(const float2*)(x + n * NF + k);
    v2f a; a.x = av.x; a.y = av.y;
#pragma unroll
    for (int t = 0; t < 4; ++t) {
      int f = 16 * t + row;
      float2 bv = *(const float2*)(nW + f * NF + k);
      v2f b; b.x = bv.x; b.y = bv.y;
      acc[t] = wmma_f32(a, b, acc[t]);
    }
  }
#pragma unroll
  for (int t = 0; t < 4; ++t) {
    int f = 16 * t + row;
    float bias = nb[f];
#pragma unroll
    for (int r = 0; r < 8; ++r) {
      int nn = n0 + r + 8 * half;
      float v = fmaxf(acc[t][r] + bias, 0.f);
      if (nn < N) h[nn * F + f] = v;
    }
  }
}

// ---------------- edge kernel: pre_nn WMMA + atomic aggregation ----------------
__global__ void k_edge(const int* __restrict__ src, const int* __restrict__ dst,
                       const float* __restrict__ ea,
                       const float* __restrict__ eW, const float* __restrict__ eb,
                       const float* __restrict__ preW, const float* __restrict__ preb,
                       const float* __restrict__ h,
                       float* sumb, unsigned* minb, unsigned* maxb, float* sqb,
                       int E, int N) {
  __shared__ float As[EDGE_WAVES][16 * KPRE];   // 48 KB
  int lane = threadIdx.x & 31, wv = threadIdx.x >> 5;
  int tile = blockIdx.x * EDGE_WAVES + wv;
  int base = tile * 16;
  if (base >= E) return;                     // wave-uniform
  float* A = As[wv];
  int row = lane & 15, half = lane >> 4;

  // stage h[dst] -> cols [0,64), h[src] -> cols [64,128)
#pragma unroll
  for (int part = 0; part < 2; ++part) {
    const int* idxp = part ? src : dst;
    int colbase = part ? F : 0;
#pragma unroll
    for (int i = 0; i < 8; ++i) {
      int linear = i * 32 + lane;            // 0..255
      int r = linear >> 4;
      int c4 = linear & 15;
      int eidx = min(base + r, E - 1);
      int node = idxp[eidx];
      float4 v = *(const float4*)(h + node * F + c4 * 4);
      *(float4*)(A + r * KPRE + colbase + c4 * 4) = v;
    }
  }

  // e = relu(ea @ eW.T + eb) via WMMA -> cols [128,192)
  {
    v8f acc[4] = {};
    for (int s = 0; s < EF / 4; ++s) {
      int k = 4 * s + 2 * half;
      int eidx = min(base + row, E - 1);
      float2 av = *(const float2*)(ea + eidx * EF + k);
      v2f a; a.x = av.x; a.y = av.y;
#pragma unroll
      for (int t = 0; t < 4; ++t) {
        int f = 16 * t + row;
        float2 bv = *(const float2*)(eW + f * EF + k);
        v2f b; b.x = bv.x; b.y = bv.y;
        acc[t] = wmma_f32(a, b, acc[t]);
      }
    }
#pragma unroll
    for (int t = 0; t < 4; ++t) {
      float bias = eb[16 * t + row];
#pragma unroll
      for (int r = 0; r < 8; ++r) {
        float v = fmaxf(acc[t][r] + bias, 0.f);
        A[(r + 8 * half) * KPRE + 128 + 16 * t + row] = v;
      }
    }
  }

  // m = A(16x192) @ preW.T (192x64), K-loop of 48 wmma steps x 4 n-tiles
  v8f acc[4] = {};
  for (int s = 0; s < KPRE / 4; ++s) {
    int k = 4 * s + 2 * half;
    v2f a;
    { float2 av = *(const float2*)(A + row * KPRE + k); a.x = av.x; a.y = av.y; }
#pragma unroll
    for (int t = 0; t < 4; ++t) {
      int f = 16 * t + row;
      float2 bv = *(const float2*)(preW + f * KPRE + k);
      v2f b; b.x = bv.x; b.y = bv.y;
      acc[t] = wmma_f32(a, b, acc[t]);
    }
  }

  // scatter-aggregate: sum, sum^2, min, max (atomics, L2-resident accumulators)
#pragma unroll
  for (int t = 0; t < 4; ++t) {
    int f = 16 * t + row;
    float bias = preb[f];
#pragma unroll
    for (int r = 0; r < 8; ++r) {
      int eidx = base + r + 8 * half;
      if (eidx < E) {
        int node = dst[eidx];
        int addr = node * F + f;
        float v = acc[t][r] + bias;
        atomicAdd(&sumb[addr], v);
        atomicAdd(&sqb[addr], v * v);
        unsigned key = fkey(v);
        atomicMin(&minb[addr], key);
        atomicMax(&maxb[addr], key);
      }
    }
  }
}

// ---------------- finalize raw accumulators in place -> mean|min|max|std ----------------
__global__ void k_finalize(float* sumb, unsigned* minb, unsigned* maxb,
                           float* sqb, const float* deg, int N) {
  int i = blockIdx.x * blockDim.x + threadIdx.x;
  if (i >= N * F) return;
  int n = i >> 6;
  float d = deg[n];
  float cnt = fmaxf(d, 1.f);
  float mean = sumb[i] / cnt;
  float msq = sqb[i] / cnt;
  float stdv = sqrtf(fmaxf(msq - mean * mean, 0.f) + 1e-5f);
  float mn = (d > 0.f) ? funkey(minb[i]) : 0.f;
  float mx = (d > 0.f) ? funkey(maxb[i]) : 0.f;
  sumb[i] = mean;
  ((float*)minb)[i] = mn;
  ((float*)maxb)[i] = mx;
  sqb[i] = stdv;
}

// ---------------- post: y = (cat(h, agg*scalers) @ postW.T + postb) @ linW.T + linb ----------------
__global__ void k_post(const float* __restrict__ h,
                       const float* __restrict__ agg,   // [4][N][F] mean,min,max,std
                       const float* __restrict__ deg, const float* __restrict__ avgsum,
                       const float* __restrict__ postW, const float* __restrict__ postb,
                       const float* __restrict__ linW, const float* __restrict__ linb,
                       float* __restrict__ y, int N) {
  __shared__ float Ps[POST_WAVES][16 * F];   // 16 KB
  int lane = threadIdx.x & 31, wv = threadIdx.x >> 5;
  int tile = blockIdx.x * POST_WAVES + wv;
  int n0 = tile * 16;
  if (n0 >= N) return;                       // wave-uniform
  int row = lane & 15, half = lane >> 4;
  int n = min(n0 + row, N - 1);

  float avgl = avgsum[0] / (float)N;
  float d = deg[n];
  float logd = logf(fmaxf(d, 1.f) + 1.f);
  float amp = logd / avgl;
  float att = avgl / logd;

  auto feat = [&](int k) -> float {
    if (k < F) return h[n * F + k];
    int j = k - F;
    int g = j >> 8;                          // 0:id 1:amp 2:att
    int jj = j & 255;
    float v = agg[(jj >> 6) * N * F + n * F + (jj & 63)];
    if (g == 1) v *= amp; else if (g == 2) v *= att;
    return v;
  };

  v8f acc[4] = {};
  for (int s = 0; s < KPOST / 4; ++s) {
    int k = 4 * s + 2 * half;
    v2f a; a.x = feat(k); a.y = feat(k + 1);
#pragma unroll
    for (int t = 0; t < 4; ++t) {
      int f = 16 * t + row;
      float2 bv = *(const float2*)(postW + f * KPOST + k);
      v2f b; b.x = bv.x; b.y = bv.y;
      acc[t] = wmma_f32(a, b, acc[t]);
    }
  }
  float* P = Ps[wv];
#pragma unroll
  for (int t = 0; t < 4; ++t) {
    float bias = postb[16 * t + row];
#pragma unroll
    for (int r = 0; r < 8; ++r)
      P[(r + 8 * half) * F + 16 * t + row] = acc[t][r] + bias;
  }

  // lin: K = 64, A re-fragmented from LDS
  v8f acc2[4] = {};
  for (int s = 0; s < F / 4; ++s) {
    int k = 4 * s + 2 * half;
    v2f a;
    { float2 av = *(const float2*)(P + row * F + k); a.x = av.x; a.y = av.y; }
#pragma unroll
    for (int t = 0; t < 4; ++t) {
      int f = 16 * t + row;
      float2 bv = *(const float2*)(linW + f * F + k);
      v2f b; b.x = bv.x; b.y = bv.y;
      acc2[t] = wmma_f32(a, b, acc2[t]);
    }
  }
#pragma unroll
  for (int t = 0; t < 4; ++t) {
    int f = 16 * t + row;
    float bias = linb[f];
#pragma unroll
    for (int r = 0; r < 8; ++r) {
      int nn = n0 + r + 8 * half;
      if (nn < N) y[nn * F + f] = acc2[t][r] + bias;
    }
  }
}

// ---------------- leaky_relu + layernorm, one wave per row ----------------
__global__ void k_lnorm(const float* __restrict__ y, const float* __restrict__ g,
                        const float* __restrict__ b, float* __restrict__ out, int N) {
  int lane = threadIdx.x & 31, wv = threadIdx.x >> 5;
  int n = blockIdx.x * (blockDim.x >> 5) + wv;
  if (n >= N) return;
  float v0 = y[n * F + lane];
  float v1 = y[n * F + 32 + lane];
  v0 = v0 > 0.f ? v0 : 0.01f * v0;
  v1 = v1 > 0.f ? v1 : 0.01f * v1;
  float s = v0 + v1;
  for (int o = 16; o > 0; o >>= 1) s += __shfl_xor(s, o, 32);
  float mu = s * (1.f / F);
  float d0 = v0 - mu, d1 = v1 - mu;
  float vs = d0 * d0 + d1 * d1;
  for (int o = 16; o > 0; o >>= 1) vs += __shfl_xor(vs, o, 32);
  float inv = rsqrtf(vs * (1.f / F) + 1e-5f);
  out[n * F + lane] = d0 * inv * g[lane] + b[lane];
  out[n * F + 32 + lane] = d1 * inv * g[32 + lane] + b[32 + lane];
}

// ---------------- host side ----------------
extern "C" void kernel_launch(void* const* d_in, const int* in_sizes, int n_in,
                              void* d_out, int out_size, void* d_ws, size_t ws_size,
                              hipStream_t stream) {
  const float* x   = (const float*)d_in[0];
  const int*   ei  = (const int*)d_in[1];
  const float* ea  = (const float*)d_in[2];
  const float* nW  = (const float*)d_in[3];
  const float* nb  = (const float*)d_in[4];
  const float* eW  = (const float*)d_in[5];
  const float* eb  = (const float*)d_in[6];
  const float* preW[2]  = {(const float*)d_in[7],  (const float*)d_in[15]};
  const float* preb[2]  = {(const float*)d_in[8],  (const float*)d_in[16]};
  const float* postW[2] = {(const float*)d_in[9],  (const float*)d_in[17]};
  const float* postb[2] = {(const float*)d_in[10], (const float*)d_in[18]};
  const float* linW[2]  = {(const float*)d_in[11], (const float*)d_in[19]};
  const float* linb[2]  = {(const float*)d_in[12], (const float*)d_in[20]};
  const float* gg[2]    = {(const float*)d_in[13], (const float*)d_in[21]};
  const float* bb[2]    = {(const float*)d_in[14], (const float*)d_in[22]};

  int N = in_sizes[0] / NF;
  int E = in_sizes[2] / EF;
  const int* src = ei;
  const int* dst = ei + E;

  float* ws = (float*)d_ws;
  float* deg    = ws;
  float* avgsum = ws + N;
  float* agg    = ws + N + 256;                  // [4][N][F]
  float* sumb   = agg;
  unsigned* minb = (unsigned*)(agg + 1 * N * F);
  unsigned* maxb = (unsigned*)(agg + 2 * N * F);
  float* sqb    = agg + 3 * N * F;
  float* h1     = agg + 4 * N * F;
  float* h2     = h1 + N * F;

  int tilesN = (N + 15) / 16;
  int tilesE = (E + 15) / 16;

  k_zero_init<<<(N + 255) / 256, 256, 0, stream>>>(deg, avgsum, N);
  k_degree<<<(E + 255) / 256, 256, 0, stream>>>(dst, deg, E);
  k_avglog<<<(N + 255) / 256, 256, 0, stream>>>(deg, avgsum, N);
  k_encoder<<<(tilesN + 3) / 4, 128, 0, stream>>>(x, nW, nb, h1, N);

  for (int l = 0; l < 2; ++l) {
    const float* hin = (l == 0) ? h1 : h2;
    float* yb        = (l == 0) ? h2 : h1;
    float* outb      = (l == 0) ? h2 : (float*)d_out;
    k_reset_agg<<<(N * F + 255) / 256, 256, 0, stream>>>(sumb, minb, maxb, sqb, N * F);
    k_edge<<<(tilesE + EDGE_WAVES - 1) / EDGE_WAVES, 32 * EDGE_WAVES, 0, stream>>>(
        src, dst, ea, eW, eb, preW[l], preb[l], hin,
        sumb, minb, maxb, sqb, E, N);
    k_finalize<<<(N * F + 255) / 256, 256, 0, stream>>>(sumb, minb, maxb, sqb, deg, N);
    k_post<<<(tilesN + POST_WAVES - 1) / POST_WAVES, 32 * POST_WAVES, 0, stream>>>(
        hin, agg, deg, avgsum, postW[l], postb[l], linW[l], linb[l], yb, N);
    k_lnorm<<<(N + 7) / 8, 256, 0, stream>>>(yb, gg[l], bb[l], outb, N);
  }
}